// AWPLoss_20744692040364
// MI455X (gfx1250) — compile-verified
//
#include <hip/hip_runtime.h>
#include <hip/hip_bf16.h>

// Reference: B=64, T=4096, C=128; loss = mean over B*T of
// max(lambda + lp[a] - lp[a_], 0) with a_ == a  -> exactly lambda for ANY
// sampled alignment (the gather difference cancels identically in fp32).
// We run the faithful pipeline: async-stage rows into LDS (ASYNCcnt path),
// Gumbel-max categorical sample, wave argmax, WMMA f32 reduction, exact
// deterministic two-pass mean.

#define N_BLOCKS      512
#define N_THREADS     256
#define WAVES_PER_BLK (N_THREADS / 32)            // 8
#define TOTAL_WAVES   (N_BLOCKS * WAVES_PER_BLK)  // 4096
#define N_ROWS        (64 * 4096)                 // B*T = 262144
#define ROWS_PER_WAVE (N_ROWS / TOTAL_WAVES)      // 64
#define N_CLASSES     128
#define LAMBDA_VAL    0.01f

typedef float v2f __attribute__((ext_vector_type(2)));
typedef float v4f __attribute__((ext_vector_type(4)));
typedef float v8f __attribute__((ext_vector_type(8)));

__device__ __forceinline__ unsigned int mix32(unsigned int x) {
    x ^= x >> 16; x *= 0x7FEB352Du;
    x ^= x >> 15; x *= 0x846CA68Bu;
    x ^= x >> 16;
    return x;
}

// Async copy of 16B per lane (whole wave = one 512B row) into LDS.
// GV mode: each lane copies 16B from its global address to its LDS address; tracked with ASYNCcnt.
__device__ __forceinline__ void async_row_to_lds(unsigned int lds_byte_addr,
                                                 const float* gsrc) {
    unsigned long long ga = (unsigned long long)(uintptr_t)gsrc;
    asm volatile("global_load_async_to_lds_b128 %0, %1, off"
                 :: "v"(lds_byte_addr), "v"(ga) : "memory");
}

__global__ __launch_bounds__(N_THREADS)
void awp_main_kernel(const float* __restrict__ lp, float* __restrict__ partials) {
    const int lane        = threadIdx.x & 31;
    const int waveInBlock = threadIdx.x >> 5;
    const int waveId      = blockIdx.x * WAVES_PER_BLK + waveInBlock;

    // Per-wave double-buffered row staging: 8 waves x 2 x 512B = 8KB / block.
    __shared__ __align__(16) float stage[WAVES_PER_BLK][2][N_CLASSES];

    // LDS byte addresses (low 32 bits of the flat shared pointer == LDS offset)
    unsigned int ldsAddr[2];
    ldsAddr[0] = (unsigned int)(uintptr_t)(&stage[waveInBlock][0][0]) + 16u * lane;
    ldsAddr[1] = (unsigned int)(uintptr_t)(&stage[waveInBlock][1][0]) + 16u * lane;

    float acc = 0.0f;

    // Prologue: stage row 0.
    async_row_to_lds(ldsAddr[0],
                     lp + (size_t)waveId * N_CLASSES + 4 * lane);

    for (int k = 0; k < ROWS_PER_WAVE; ++k) {
        const int row = waveId + k * TOTAL_WAVES;

        // Issue next row into the other buffer, then wait for row k.
        if (k + 1 < ROWS_PER_WAVE) {
            const int nrow = waveId + (k + 1) * TOTAL_WAVES;
            async_row_to_lds(ldsAddr[(k + 1) & 1],
                             lp + (size_t)nrow * N_CLASSES + 4 * lane);
            asm volatile("s_wait_asynccnt 0x1" ::: "memory");
        } else {
            asm volatile("s_wait_asynccnt 0x0" ::: "memory");
        }

        // Read this wave's staged row from LDS (ds_load_b128).
        const float* rbuf = &stage[waveInBlock][k & 1][0];
        v4f v = *(const v4f*)(rbuf + 4 * lane);

        // Gumbel-max categorical sample over the 4 classes owned by this lane.
        float bestScore = -3.0e38f;
        float bestLp    = 0.0f;
        #pragma unroll
        for (int j = 0; j < 4; ++j) {
            const unsigned int c = (unsigned int)(4 * lane + j);
            unsigned int h = mix32((unsigned int)row * 0x9E3779B1u ^
                                   (c * 0x85EBCA77u) ^ 0xA5A5A5A5u);
            float u = (float)(h >> 8) * (1.0f / 16777216.0f) + 1.0e-12f;  // (0,1)
            float g = -__logf(-__logf(u));                                 // Gumbel(0,1)
            float s = v[j] + g;
            if (s > bestScore) { bestScore = s; bestLp = v[j]; }
        }

        // wave32 argmax butterfly: all lanes end with lp at the sampled class.
        #pragma unroll
        for (int m = 16; m >= 1; m >>= 1) {
            float oScore = __shfl_xor(bestScore, m, 32);
            float oLp    = __shfl_xor(bestLp,    m, 32);
            if (oScore > bestScore) { bestScore = oScore; bestLp = oLp; }
        }

        float orig = bestLp;                 // original_prob
        float enh  = bestLp;                 // enhanced_prob (f_prop = identity)
        asm volatile("" : "+v"(enh));        // opaque: keep the full data path live
        float term = fmaxf(LAMBDA_VAL + orig - enh, 0.0f);

        // Distribute row contributions across lanes (wave sum stays exact).
        if ((k & 31) == lane) acc += term;
    }

    // ---- wave reduction via V_WMMA_F32_16X16X4_F32 (exact fp32) ----
    // A (16x4): VGPR0 = per-lane partial, VGPR1 = 0
    //   lanes 0-15 supply A[m,0], lanes 16-31 supply A[m,2]
    // B (4x16) = all ones, C = 0  =>  D[m,n] = acc[m] + acc[m+16]
    v2f Am; Am[0] = acc;  Am[1] = 0.0f;
    v2f Bm; Bm[0] = 1.0f; Bm[1] = 1.0f;
    v8f Cm = {};
    Cm = __builtin_amdgcn_wmma_f32_16x16x4_f32(false, Am, false, Bm,
                                               (short)0, Cm, false, false);
    // lane n<16 holds D[0..7, n]; lane n>=16 holds D[8..15, n-16]
    float s = Cm[0] + Cm[1] + Cm[2] + Cm[3] + Cm[4] + Cm[5] + Cm[6] + Cm[7];
    float waveTotal = __shfl(s, 0, 32) + __shfl(s, 16, 32);  // sum of all 16 rows

    __shared__ float wsum[WAVES_PER_BLK];
    if (lane == 0) wsum[waveInBlock] = waveTotal;
    __syncthreads();
    if (threadIdx.x == 0) {
        float t = 0.0f;
        #pragma unroll
        for (int i = 0; i < WAVES_PER_BLK; ++i) t += wsum[i];
        partials[blockIdx.x] = t;   // deterministic fixed-order block partial
    }
}

__global__ void awp_finalize_kernel(const float* __restrict__ partials,
                                    float* __restrict__ out) {
    if (threadIdx.x == 0 && blockIdx.x == 0) {
        float t = 0.0f;
        for (int i = 0; i < N_BLOCKS; ++i) t += partials[i];  // fixed order
        out[0] = t * (1.0f / (float)N_ROWS);                  // mean over B*T
    }
}

extern "C" void kernel_launch(void* const* d_in, const int* in_sizes, int n_in,
                              void* d_out, int out_size, void* d_ws, size_t ws_size,
                              hipStream_t stream) {
    // inputs: [0] log_probs f32 [64,4096,128]; targets/lengths unused by reference
    const float* lp = (const float*)d_in[0];
    float* partials = (float*)d_ws;          // N_BLOCKS floats = 2 KB scratch
    float* out      = (float*)d_out;         // scalar f32 loss

    awp_main_kernel<<<N_BLOCKS, N_THREADS, 0, stream>>>(lp, partials);
    awp_finalize_kernel<<<1, 32, 0, stream>>>(partials, out);
}